// FocusedLinearAttention_90512140796374
// MI455X (gfx1250) — compile-verified
//
#include <hip/hip_runtime.h>

typedef __attribute__((ext_vector_type(2))) float v2f;
typedef __attribute__((ext_vector_type(4))) float f4;
typedef __attribute__((ext_vector_type(8))) float v8f;

namespace {
constexpr int IMG   = 64;
constexpr int NPIX  = IMG * IMG;   // N = 4096
constexpr int CH    = 512;         // C
constexpr int C2    = 2 * CH;      // 1024
constexpr int HD    = 32;          // head_dim
constexpr int NHEAD = CH / HD;     // 16
constexpr int KMAX  = CH / 4;      // 128 (rope frequencies)
}

// ---------------------------------------------------------------------------
// K1: qk = x @ W_qk^T + b_qk   (M=B*N, N=1024, K=512), fp32 WMMA 16x16x4
// Block: 128 threads (4 waves), tile 64x64, K-step 16.
// ---------------------------------------------------------------------------
__global__ __launch_bounds__(128)
void k_qk_gemm(const float* __restrict__ x, const float* __restrict__ Wqk,
               const float* __restrict__ bqk, float* __restrict__ qk)
{
    __shared__ float As[16][65];   // As[k][m]
    __shared__ float Bs[16][65];   // Bs[k][n]
    const int tid  = threadIdx.x;
    const int lane = tid & 31;
    const int wave = tid >> 5;
    const int l16  = lane & 15;
    const int hi   = lane >> 4;          // 0 for lanes 0-15, 1 for 16-31
    const int wm   = (wave & 1) * 32;    // wave M offset within tile
    const int wn   = (wave >> 1) * 32;   // wave N offset within tile
    const long row0 = (long)blockIdx.y * 64;
    const int  col0 = blockIdx.x * 64;

    v8f acc[2][2] = {};

    for (int k0 = 0; k0 < CH; k0 += 16) {
        // A tile: As[k][m] = x[row0+m][k0+k]
        {
            int m  = tid >> 1;
            int kq = (tid & 1) * 8;
            const float* src = x + (row0 + m) * CH + k0 + kq;
            f4 a0 = *(const f4*)(src);
            f4 a1 = *(const f4*)(src + 4);
            #pragma unroll
            for (int j = 0; j < 4; ++j) { As[kq + j][m] = a0[j]; As[kq + 4 + j][m] = a1[j]; }
        }
        // B tile: Bs[k][n] = Wqk[col0+n][k0+k]   (B = W^T, i.e. KxN)
        {
            int n  = tid >> 1;
            int kq = (tid & 1) * 8;
            const float* src = Wqk + (long)(col0 + n) * CH + k0 + kq;
            f4 b0 = *(const f4*)(src);
            f4 b1 = *(const f4*)(src + 4);
            #pragma unroll
            for (int j = 0; j < 4; ++j) { Bs[kq + j][n] = b0[j]; Bs[kq + 4 + j][n] = b1[j]; }
        }
        __syncthreads();

        #pragma unroll
        for (int kk = 0; kk < 16; kk += 4) {
            const int ke = kk + (hi ? 2 : 0);   // even vgpr K index for this lane half
            const int ko = kk + (hi ? 3 : 1);   // odd  vgpr K index
            v2f a0, a1, b0, b1;
            a0.x = As[ke][wm + l16];       a0.y = As[ko][wm + l16];
            a1.x = As[ke][wm + 16 + l16];  a1.y = As[ko][wm + 16 + l16];
            b0.x = Bs[ke][wn + l16];       b0.y = Bs[ko][wn + l16];
            b1.x = Bs[ke][wn + 16 + l16];  b1.y = Bs[ko][wn + 16 + l16];
            acc[0][0] = __builtin_amdgcn_wmma_f32_16x16x4_f32(false, a0, false, b0, (short)0, acc[0][0], false, false);
            acc[0][1] = __builtin_amdgcn_wmma_f32_16x16x4_f32(false, a0, false, b1, (short)0, acc[0][1], false, false);
            acc[1][0] = __builtin_amdgcn_wmma_f32_16x16x4_f32(false, a1, false, b0, (short)0, acc[1][0], false, false);
            acc[1][1] = __builtin_amdgcn_wmma_f32_16x16x4_f32(false, a1, false, b1, (short)0, acc[1][1], false, false);
        }
        __syncthreads();
    }

    // C layout: vgpr r -> M = r + 8*hi, N = l16
    #pragma unroll
    for (int i = 0; i < 2; ++i) {
        #pragma unroll
        for (int j = 0; j < 2; ++j) {
            const int col = col0 + wn + j * 16 + l16;
            const float bias = bqk[col];
            #pragma unroll
            for (int r = 0; r < 8; ++r) {
                const long row = row0 + wm + i * 16 + r + hi * 8;
                qk[row * C2 + col] = acc[i][j][r] + bias;
            }
        }
    }
}

// ---------------------------------------------------------------------------
// K2: per-row RoPE + elu+1 + /softplus(scale) + focusing (f=3), in place.
// One block per (b,n) row; thread t owns q-pair t and k-pair t (256 pairs).
// ---------------------------------------------------------------------------
__global__ __launch_bounds__(256)
void k_focus(float* __restrict__ qk, const float* __restrict__ scale_p)
{
    __shared__ float red[4][256];
    const int row = blockIdx.x;            // b*N + n
    const int n   = row & (NPIX - 1);
    const float hh = (float)(n >> 6);
    const float ww = (float)(n & 63);
    const int t = threadIdx.x;             // pair index 0..255
    float* qrow = qk + (long)row * C2;
    float* krow = qrow + CH;

    const int pm = t & (KMAX - 1);
    const float theta = __expf(-(float)pm * (9.210340371976184f / (float)KMAX)); // 10000^(-pm/128)
    const float pos = (t < KMAX) ? hh : ww;
    float sn, cs;
    __sincosf(pos * theta, &sn, &cs);

    const int cre = 2 * t, cim = 2 * t + 1;
    const float sc_re = __logf(1.f + __expf(scale_p[cre]));   // softplus
    const float sc_im = __logf(1.f + __expf(scale_p[cim]));

    float q_re = qrow[cre], q_im = qrow[cim];
    float k_re = krow[cre], k_im = krow[cim];

    // RoPE
    float tq = q_re * cs - q_im * sn; q_im = q_im * cs + q_re * sn; q_re = tq;
    float tk = k_re * cs - k_im * sn; k_im = k_im * cs + k_re * sn; k_re = tk;
    // elu + 1
    q_re = (q_re > 0.f) ? q_re + 1.f : __expf(q_re);
    q_im = (q_im > 0.f) ? q_im + 1.f : __expf(q_im);
    k_re = (k_re > 0.f) ? k_re + 1.f : __expf(k_re);
    k_im = (k_im > 0.f) ? k_im + 1.f : __expf(k_im);
    // / softplus(scale)
    q_re /= sc_re; q_im /= sc_im; k_re /= sc_re; k_im /= sc_im;

    const float q3_re = q_re * q_re * q_re, q3_im = q_im * q_im * q_im;
    const float k3_re = k_re * k_re * k_re, k3_im = k_im * k_im * k_im;

    red[0][t] = q_re * q_re + q_im * q_im;
    red[1][t] = q3_re * q3_re + q3_im * q3_im;
    red[2][t] = k_re * k_re + k_im * k_im;
    red[3][t] = k3_re * k3_re + k3_im * k3_im;
    __syncthreads();
    for (int s = 128; s > 0; s >>= 1) {
        if (t < s) {
            red[0][t] += red[0][t + s];
            red[1][t] += red[1][t + s];
            red[2][t] += red[2][t + s];
            red[3][t] += red[3][t + s];
        }
        __syncthreads();
    }
    const float fq = sqrtf(red[0][0]) * rsqrtf(red[1][0]);  // ||q|| / ||q^3||
    const float fk = sqrtf(red[2][0]) * rsqrtf(red[3][0]);

    qrow[cre] = q3_re * fq;  qrow[cim] = q3_im * fq;
    krow[cre] = k3_re * fk;  krow[cim] = k3_im * fk;
}

// ---------------------------------------------------------------------------
// K3a/K3b: k_mean over N (two-stage, deterministic, no atomics)
// ---------------------------------------------------------------------------
__global__ __launch_bounds__(256)
void k_ksum_partial(const float* __restrict__ qk, float* __restrict__ partial)
{
    const int b = blockIdx.x >> 4;
    const int chunk = blockIdx.x & 15;
    const int t = threadIdx.x;
    const float* base = qk + ((long)(b * NPIX + chunk * 256)) * C2 + CH;
    float a0 = 0.f, a1 = 0.f;
    for (int r = 0; r < 256; ++r) {
        a0 += base[(long)r * C2 + t];
        a1 += base[(long)r * C2 + t + 256];
    }
    partial[(long)blockIdx.x * CH + t]       = a0;
    partial[(long)blockIdx.x * CH + t + 256] = a1;
}

__global__ __launch_bounds__(256)
void k_kmean(const float* __restrict__ partial, float* __restrict__ kmean, int B)
{
    const int idx = blockIdx.x * 256 + threadIdx.x;  // b*CH + c
    if (idx >= B * CH) return;
    const int b = idx / CH, c = idx - b * CH;
    float s = 0.f;
    for (int ch = 0; ch < 16; ++ch) s += partial[((long)(b * 16 + ch)) * CH + c];
    kmean[idx] = s * (1.f / (float)NPIX);
}

// ---------------------------------------------------------------------------
// K4: kv[b,h] = (1/N) * k^T v   (32x32, K=4096). LDS-tiled FMA (tiny GEMM).
// ---------------------------------------------------------------------------
__global__ __launch_bounds__(256)
void k_kv(const float* __restrict__ qk, const float* __restrict__ x,
          float* __restrict__ kv)
{
    __shared__ float kt[64][33];
    __shared__ float vt[64][33];
    const int bh = blockIdx.x;
    const int b = bh / NHEAD, h = bh - b * NHEAD;
    const int t = threadIdx.x;
    const int dk  = t >> 3;           // 0..31
    const int dv0 = (t & 7) * 4;      // 0,4,...,28
    float acc[4] = {0.f, 0.f, 0.f, 0.f};

    for (int n0 = 0; n0 < NPIX; n0 += 64) {
        #pragma unroll
        for (int i = 0; i < 8; ++i) {
            const int flat = t + i * 256;
            const int r = flat >> 5, c = flat & 31;
            const long nrow = (long)(b * NPIX + n0 + r);
            kt[r][c] = qk[nrow * C2 + CH + h * HD + c];
            vt[r][c] = x[nrow * CH + h * HD + c];
        }
        __syncthreads();
        for (int i = 0; i < 64; ++i) {
            const float kval = kt[i][dk];
            #pragma unroll
            for (int j = 0; j < 4; ++j) acc[j] += kval * vt[i][dv0 + j];
        }
        __syncthreads();
    }
    #pragma unroll
    for (int j = 0; j < 4; ++j)
        kv[((long)bh * HD + dk) * HD + dv0 + j] = acc[j] * (1.f / (float)NPIX);
}

// ---------------------------------------------------------------------------
// K5: out = (q @ kv) * z,  z = 1/(q . k_mean + 1e-6).  One wave per 16-row tile.
// WMMA f32 16x16x4, K=32 (d), two N-halves of 16 (dv).
// ---------------------------------------------------------------------------
__global__ __launch_bounds__(32)
void k_out(const float* __restrict__ qk, const float* __restrict__ kvbuf,
           const float* __restrict__ kmean, float* __restrict__ out)
{
    __shared__ float qs[16][33];
    __shared__ float kvs[32][33];
    __shared__ float kms[32];
    __shared__ float zs[16];
    const int bh = blockIdx.y;
    const int b = bh / NHEAD, h = bh - b * NHEAD;
    const int n0 = blockIdx.x * 16;
    const int lane = threadIdx.x;
    const int l16 = lane & 15;
    const int hi  = lane >> 4;

    for (int i = 0; i < 16; ++i)
        qs[i][lane] = qk[((long)(b * NPIX + n0 + i)) * C2 + h * HD + lane];
    #pragma unroll
    for (int i = 0; i < 32; ++i)
        kvs[i][lane] = kvbuf[((long)bh * HD + i) * HD + lane];
    kms[lane] = kmean[b * CH + h * HD + lane];
    __syncthreads();

    if (lane < 16) {
        float s = 0.f;
        for (int dd = 0; dd < HD; ++dd) s += qs[lane][dd] * kms[dd];
        zs[lane] = 1.f / (s + 1e-6f);
    }
    __syncthreads();

    v8f acc0 = {}, acc1 = {};
    #pragma unroll
    for (int kk = 0; kk < HD; kk += 4) {
        const int ke = kk + (hi ? 2 : 0);
        const int ko = kk + (hi ? 3 : 1);
        v2f a, b0, b1;
        a.x  = qs[l16][ke];        a.y  = qs[l16][ko];
        b0.x = kvs[ke][l16];       b0.y = kvs[ko][l16];
        b1.x = kvs[ke][16 + l16];  b1.y = kvs[ko][16 + l16];
        acc0 = __builtin_amdgcn_wmma_f32_16x16x4_f32(false, a, false, b0, (short)0, acc0, false, false);
        acc1 = __builtin_amdgcn_wmma_f32_16x16x4_f32(false, a, false, b1, (short)0, acc1, false, false);
    }

    #pragma unroll
    for (int r = 0; r < 8; ++r) {
        const int m = r + hi * 8;
        const long orow = ((long)(b * NPIX + n0 + m)) * CH + h * HD;
        const float z = zs[m];
        out[orow + l16]      = acc0[r] * z;
        out[orow + 16 + l16] = acc1[r] * z;
    }
}

// ---------------------------------------------------------------------------
// K6: depthwise 5x5 conv of v (= x, NHWC with C innermost) added into out.
// c innermost across threads -> coalesced; x is L2-resident (64MB < 192MB L2).
// ---------------------------------------------------------------------------
__global__ __launch_bounds__(256)
void k_conv_add(const float* __restrict__ x, const float* __restrict__ w,
                const float* __restrict__ bias, float* __restrict__ out, int total)
{
    const int idx = blockIdx.x * 256 + threadIdx.x;
    if (idx >= total) return;
    const int c = idx & (CH - 1);
    const int n = (idx >> 9) & (NPIX - 1);
    const int b = idx >> 21;            // / (CH*NPIX) = 2^21
    const int d = c & (HD - 1);
    const int hh = n >> 6, ww = n & 63;
    float acc = bias[d];
    #pragma unroll
    for (int dy = -2; dy <= 2; ++dy) {
        const int yy = hh + dy;
        if (yy < 0 || yy >= IMG) continue;
        #pragma unroll
        for (int dx = -2; dx <= 2; ++dx) {
            const int xx = ww + dx;
            if (xx < 0 || xx >= IMG) continue;
            acc += x[((long)b * NPIX + yy * IMG + xx) * CH + c] *
                   w[d * 25 + (dy + 2) * 5 + (dx + 2)];
        }
    }
    out[idx] += acc;
}

// ---------------------------------------------------------------------------
extern "C" void kernel_launch(void* const* d_in, const int* in_sizes, int n_in,
                              void* d_out, int out_size, void* d_ws, size_t ws_size,
                              hipStream_t stream)
{
    const float* x      = (const float*)d_in[0];
    const float* Wqk    = (const float*)d_in[1];
    const float* bqk    = (const float*)d_in[2];
    const float* scalep = (const float*)d_in[3];
    const float* dwcw   = (const float*)d_in[4];
    const float* dwcb   = (const float*)d_in[5];
    // d_in[6]/d_in[7] are img_H/img_W (= 64, fixed by setup_inputs)

    const int  B    = in_sizes[0] / (NPIX * CH);
    const long Mtot = (long)B * NPIX;

    char* ws = (char*)d_ws;
    size_t o = 0;
    float* qk      = (float*)(ws + o);  o += (size_t)Mtot * C2 * sizeof(float);        // 128 MB
    float* partial = (float*)(ws + o);  o += (size_t)B * 16 * CH * sizeof(float);
    float* kmean   = (float*)(ws + o);  o += (size_t)B * CH * sizeof(float);
    float* kvbuf   = (float*)(ws + o);  // B*NHEAD*HD*HD floats

    float* out = (float*)d_out;

    k_qk_gemm<<<dim3(C2 / 64, (unsigned)(Mtot / 64)), 128, 0, stream>>>(x, Wqk, bqk, qk);
    k_focus<<<dim3((unsigned)Mtot), 256, 0, stream>>>(qk, scalep);
    k_ksum_partial<<<dim3(B * 16), 256, 0, stream>>>(qk, partial);
    k_kmean<<<dim3((B * CH + 255) / 256), 256, 0, stream>>>(partial, kmean, B);
    k_kv<<<dim3(B * NHEAD), 256, 0, stream>>>(qk, x, kvbuf);
    k_out<<<dim3(NPIX / 16, B * NHEAD), 32, 0, stream>>>(qk, kvbuf, kmean, out);
    const int total = B * NPIX * CH;
    k_conv_add<<<dim3((total + 255) / 256), 256, 0, stream>>>(x, dwcw, dwcb, out, total);
}